// HAN1_47047071760701
// MI455X (gfx1250) — compile-verified
//
#include <hip/hip_runtime.h>
#include <hip/hip_bf16.h>

#define N_NODES 4096
#define FIN     512
#define HID     256
#define HEADS   4
#define DHEAD   256
#define FOUT    (HEADS*DHEAD)   // 1024
#define TOPK    29
#define NWORDS  (N_NODES/32)    // 128
#define CAP     512             // max sources kept per destination (expected ~29)

typedef float v2f __attribute__((ext_vector_type(2)));
typedef float v8f __attribute__((ext_vector_type(8)));

// ---------------------------------------------------------------------------
// fp32 WMMA GEMM: C[M x N] = act(A[M x K] @ B[K x N]).
// Each wave owns a 16(M) x 64(N) strip: one A frag (float2 vector load) is
// reused by 4 back-to-back V_WMMA_F32_16X16X4_F32 into 4 accumulators,
// maximizing wmma density per loaded byte on the CDNA5 matrix pipe.
// A frag (16x4): lanes 0-15 hold M=lane, K={0,1}; lanes 16-31 hold M=lane-16, K={2,3}
// B frag (4x16): lanes 0-15 hold N=lane, K={0,1}; lanes 16-31 hold N=lane-16, K={2,3}
// C/D (16x16):   VGPR r -> M=r (lanes 0-15) / M=r+8 (lanes 16-31), N=lane&15
// ---------------------------------------------------------------------------
__global__ __launch_bounds__(256)
void gemm_f32_wmma(const float* __restrict__ A, const float* __restrict__ B,
                   float* __restrict__ C, int M, int N, int K, int act)
{
    const int wave = threadIdx.x >> 5;          // 0..7
    const int lane = threadIdx.x & 31;
    const int mt = blockIdx.y * 8 + wave;       // 16-row tile index
    const int m0 = mt * 16;
    const int n0 = blockIdx.x * 64;             // 64-wide N strip
    const int lr = lane & 15;
    const int hk = (lane >> 4) * 2;             // K sub-offset: 0 or 2

    const float* __restrict__ Arow  = A + (size_t)(m0 + lr) * K + hk;   // 8B aligned (K even)
    const float* __restrict__ Bbase = B + (size_t)n0 + lr;

    v8f acc0 = {}, acc1 = {}, acc2 = {}, acc3 = {};
    for (int k = 0; k < K; k += 4) {
        // prefetch A 16 floats (four K-steps) ahead -> global_prefetch_b8
        __builtin_prefetch(Arow + k + 16, 0, 1);

        const v2f a = *(const v2f*)(Arow + k);  // A[m][k+hk], A[m][k+hk+1]

        const float* __restrict__ Bk0 = Bbase + (size_t)(k + hk) * N;
        const float* __restrict__ Bk1 = Bbase + (size_t)(k + hk + 1) * N;
        v2f b0, b1, b2, b3;
        b0.x = Bk0[0];   b0.y = Bk1[0];
        b1.x = Bk0[16];  b1.y = Bk1[16];
        b2.x = Bk0[32];  b2.y = Bk1[32];
        b3.x = Bk0[48];  b3.y = Bk1[48];

        acc0 = __builtin_amdgcn_wmma_f32_16x16x4_f32(false, a, false, b0, (short)0, acc0, false, false);
        acc1 = __builtin_amdgcn_wmma_f32_16x16x4_f32(false, a, false, b1, (short)0, acc1, false, false);
        acc2 = __builtin_amdgcn_wmma_f32_16x16x4_f32(false, a, false, b2, (short)0, acc2, false, false);
        acc3 = __builtin_amdgcn_wmma_f32_16x16x4_f32(false, a, false, b3, (short)0, acc3, false, false);
    }

    const int ml = (lane >> 4) * 8;
    #pragma unroll
    for (int r = 0; r < 8; ++r) {
        float v0 = acc0[r], v1 = acc1[r], v2 = acc2[r], v3 = acc3[r];
        if (act) {
            v0 = v0 > 0.f ? v0 : 0.01f * v0;    // leaky_relu(0.01)
            v1 = v1 > 0.f ? v1 : 0.01f * v1;
            v2 = v2 > 0.f ? v2 : 0.01f * v2;
            v3 = v3 > 0.f ? v3 : 0.01f * v3;
        }
        float* crow = C + (size_t)(m0 + ml + r) * N + n0 + lr;
        crow[0]  = v0;
        crow[16] = v1;
        crow[32] = v2;
        crow[48] = v3;
    }
}

// ---------------------------------------------------------------------------
// el[n,h] = dot(feat[n,h,:], attn_l[h,:]);  er likewise. One block per node.
// ---------------------------------------------------------------------------
__global__ __launch_bounds__(256)
void elr_kernel(const float* __restrict__ feat,
                const float* __restrict__ attn_l, const float* __restrict__ attn_r,
                float* __restrict__ el, float* __restrict__ er)
{
    __shared__ float rl[256], rr[256];
    const int n = blockIdx.x, t = threadIdx.x;
    for (int h = 0; h < HEADS; ++h) {
        const float v = feat[(size_t)n * FOUT + h * DHEAD + t];
        rl[t] = v * attn_l[h * DHEAD + t];
        rr[t] = v * attn_r[h * DHEAD + t];
        __syncthreads();
        for (int s = 128; s > 0; s >>= 1) {
            if (t < s) { rl[t] += rl[t + s]; rr[t] += rr[t + s]; }
            __syncthreads();
        }
        if (t == 0) { el[n * HEADS + h] = rl[0]; er[n * HEADS + h] = rr[0]; }
        __syncthreads();
    }
}

// ---------------------------------------------------------------------------
// Per-row top-29 of filt = lrelu01(adj*simlar)*(adj>0), emitted as a bitmask.
// Zeros outrank negative lrelu values in top_k and then fail (newmat != 0),
// so mask == { filt > 0  AND  filt among the top-29 positive values }.
// Threshold found by binary search on value (48 iters -> sub-ulp interval).
// ---------------------------------------------------------------------------
__global__ __launch_bounds__(256)
void topk_mask_kernel(const float* __restrict__ simlar, const float* __restrict__ adj,
                      unsigned* __restrict__ maskw)
{
    __shared__ float rowv[N_NODES];   // 16 KB
    __shared__ float redf[256];
    __shared__ int   redi[256];
    __shared__ float sh_lo, sh_hi;
    __shared__ int   sh_K;

    const int s = blockIdx.x, t = threadIdx.x;

    float lmax = 0.f; int lpos = 0;
    for (int j = t; j < N_NODES; j += 256) {
        const float a  = adj[(size_t)s * N_NODES + j];
        const float si = simlar[(size_t)s * N_NODES + j];
        const float p  = a * si;
        const float f  = (a > 0.f) ? (p > 0.f ? p : 0.01f * p) : 0.f;
        rowv[j] = f;
        if (f > 0.f) { ++lpos; lmax = fmaxf(lmax, f); }
    }
    redf[t] = lmax; redi[t] = lpos;
    __syncthreads();
    for (int st = 128; st > 0; st >>= 1) {
        if (t < st) { redf[t] = fmaxf(redf[t], redf[t + st]); redi[t] += redi[t + st]; }
        __syncthreads();
    }
    if (t == 0) {
        const int npos = redi[0];
        sh_K  = npos < TOPK ? npos : TOPK;
        sh_lo = 0.f;
        sh_hi = redf[0] * 1.0000002f + 1e-30f;
    }
    __syncthreads();

    const int K = sh_K;
    float thr = 0.f;
    if (K > 0) {
        for (int it = 0; it < 48; ++it) {
            const float mid = 0.5f * (sh_lo + sh_hi);
            int c = 0;
            for (int j = t; j < N_NODES; j += 256) {
                const float f = rowv[j];
                c += (f > 0.f && f >= mid) ? 1 : 0;
            }
            redi[t] = c;
            __syncthreads();
            for (int st = 128; st > 0; st >>= 1) {
                if (t < st) redi[t] += redi[t + st];
                __syncthreads();
            }
            if (t == 0) { if (redi[0] >= K) sh_lo = mid; else sh_hi = mid; }
            __syncthreads();
        }
        thr = sh_lo;
    }

    for (int w = t; w < NWORDS; w += 256) {
        unsigned bits = 0u;
        if (K > 0) {
            #pragma unroll 4
            for (int b = 0; b < 32; ++b) {
                const float f = rowv[w * 32 + b];
                if (f > 0.f && f >= thr) bits |= (1u << b);
            }
        }
        maskw[(size_t)s * NWORDS + w] = bits;
    }
}

// ---------------------------------------------------------------------------
// Per-destination masked softmax over sources + aggregation + ELU.
// Stream-compaction (ordered by s) keeps every run's summation order fixed.
// out[d, h*256+f] = elu( sum_s alpha[s,d,h]*feat[s,h,f] + bias[h*256+f] )
// ---------------------------------------------------------------------------
__global__ __launch_bounds__(256)
void agg_kernel(const unsigned* __restrict__ maskw,
                const float* __restrict__ el, const float* __restrict__ er,
                const float* __restrict__ feat, const float* __restrict__ bias,
                float* __restrict__ out)
{
    __shared__ int   src[CAP];
    __shared__ float ew[CAP * HEADS];    // e logits, then alpha weights in-place
    __shared__ int   cnts[256];
    __shared__ float mh[HEADS], dh[HEADS], erd[HEADS];

    const int d = blockIdx.x, t = threadIdx.x;
    if (t < HEADS) erd[t] = er[d * HEADS + t];

    // ordered compaction of sources: thread t owns s in [t*16, t*16+16)
    const int wword = d >> 5;
    const unsigned bit = 1u << (d & 31);
    int local[16]; int lc = 0;
    #pragma unroll
    for (int i = 0; i < 16; ++i) {
        const int s = t * 16 + i;
        if (maskw[(size_t)s * NWORDS + wword] & bit) local[lc++] = s;
    }
    cnts[t] = lc;
    __syncthreads();
    for (int off = 1; off < 256; off <<= 1) {       // inclusive scan
        const int v   = cnts[t];
        const int add = (t >= off) ? cnts[t - off] : 0;
        __syncthreads();
        cnts[t] = v + add;
        __syncthreads();
    }
    const int excl  = cnts[t] - lc;
    const int total = cnts[255];
    const int cnt   = total < CAP ? total : CAP;
    for (int i = 0; i < lc; ++i) {
        const int pos = excl + i;
        if (pos < CAP) src[pos] = local[i];
    }
    __syncthreads();

    // e[s,d,h] = leaky_relu(el[s,h] + er[d,h], 0.2)
    for (int i = t; i < cnt; i += 256) {
        const int s = src[i];
        #pragma unroll
        for (int h = 0; h < HEADS; ++h) {
            float e = el[s * HEADS + h] + erd[h];
            ew[i * HEADS + h] = e > 0.f ? e : 0.2f * e;
        }
    }
    __syncthreads();

    if (t < HEADS) {                       // per-head max & denom (fixed order)
        float m = -1e30f;
        for (int i = 0; i < cnt; ++i) m = fmaxf(m, ew[i * HEADS + t]);
        float den = 0.f;
        for (int i = 0; i < cnt; ++i) den += expf(ew[i * HEADS + t] - m);
        mh[t] = m; dh[t] = fmaxf(den, 1e-20f);
    }
    __syncthreads();

    for (int i = t; i < cnt; i += 256) {   // alpha
        #pragma unroll
        for (int h = 0; h < HEADS; ++h)
            ew[i * HEADS + h] = expf(ew[i * HEADS + h] - mh[h]) / dh[h];
    }
    __syncthreads();

    // accumulate: f = t, coalesced reads of feat rows (feat fits in 192MB L2)
    #pragma unroll
    for (int h = 0; h < HEADS; ++h) {
        float acc = 0.f;
        for (int i = 0; i < cnt; ++i)
            acc += ew[i * HEADS + h] * feat[(size_t)src[i] * FOUT + h * DHEAD + t];
        const float o = acc + bias[h * DHEAD + t];
        out[(size_t)d * FOUT + h * DHEAD + t] = o > 0.f ? o : expm1f(o);
    }
}

// ---------------------------------------------------------------------------
extern "C" void kernel_launch(void* const* d_in, const int* in_sizes, int n_in,
                              void* d_out, int out_size, void* d_ws, size_t ws_size,
                              hipStream_t stream)
{
    const float* h        = (const float*)d_in[0];
    const float* simlar   = (const float*)d_in[1];
    const float* adj      = (const float*)d_in[2];
    const float* W_trans  = (const float*)d_in[3];
    const float* gat_W    = (const float*)d_in[4];
    const float* attn_l   = (const float*)d_in[5];
    const float* attn_r   = (const float*)d_in[6];
    const float* gat_bias = (const float*)d_in[7];
    // d_in[8..10] (sem_W1, sem_b1, sem_W2): softmax over a single meta-path is
    // identically 1.0, so out == sem; the semantic-attention MLP cancels out.

    float* out = (float*)d_out;
    char*  ws  = (char*)d_ws;
    float*    x     = (float*)(ws);                                   //  4 MB
    float*    feat  = (float*)(ws + ((size_t)4  << 20));              // 16 MB
    float*    el    = (float*)(ws + ((size_t)20 << 20));              // 64 KB
    float*    er    = (float*)(ws + ((size_t)20 << 20) + (64 << 10)); // 64 KB
    unsigned* maskw = (unsigned*)(ws + ((size_t)20 << 20) + (128 << 10)); // 2 MB

    const dim3 blk(256);

    // x = lrelu01(h @ W_trans)        [4096x512 @ 512x256], 64-wide N strips
    gemm_f32_wmma<<<dim3(HID / 64, N_NODES / 128), blk, 0, stream>>>(
        h, W_trans, x, N_NODES, HID, FIN, /*act=*/1);

    // feat = x @ gat_W                [4096x256 @ 256x1024]
    gemm_f32_wmma<<<dim3(FOUT / 64, N_NODES / 128), blk, 0, stream>>>(
        x, gat_W, feat, N_NODES, FOUT, HID, /*act=*/0);

    // el, er
    elr_kernel<<<dim3(N_NODES), blk, 0, stream>>>(feat, attn_l, attn_r, el, er);

    // top-29 mask bits
    topk_mask_kernel<<<dim3(N_NODES), blk, 0, stream>>>(simlar, adj, maskw);

    // masked softmax aggregation + ELU -> out
    agg_kernel<<<dim3(N_NODES), blk, 0, stream>>>(maskw, el, er, feat, gat_bias, out);
}